// NLL_20203526160574
// MI455X (gfx1250) — compile-verified
//
#include <hip/hip_runtime.h>
#include <hip/hip_bf16.h>
#include <math.h>

// ---------------------------------------------------------------------------
// Problem constants (match reference): x (B,T,H,W) = (4,8,256,256) f32,
// conv_w (64,8,3,3) f32, scalar f32 output.
// ---------------------------------------------------------------------------
#define Bn 4
#define Tn 8
#define Hn 256
#define Wn 256
#define NP (Bn*Tn*Hn*Wn)        // 2,097,152 elements per (B,T,H,W) plane-set
#define NSEG (Wn/32)            // 8 row segments of 32 pixels
#define NBLK (Bn*Hn*NSEG)       // 8192 blocks
#define KTOT 72                 // im2col K = 8 in-ch * 3 * 3

typedef float v2f __attribute__((ext_vector_type(2)));
typedef float v8f __attribute__((ext_vector_type(8)));

__device__ __forceinline__ float fsigmoid(float v) { return 1.0f / (1.0f + __expf(-v)); }
__device__ __forceinline__ float csig(float v, float mn, float mx) { return fsigmoid(v) * (mx - mn) + mn; }

// ---------------------------------------------------------------------------
// Kernel 1: 3x3 SAME conv as implicit GEMM with V_WMMA_F32_16X16X4_F32.
// Inner loop uses only immediate-offset LDS loads:
//   A: ds_load_b64 from lw at (step*16) bytes past a fixed per-lane base
//   B: two ds_load_b32 from im2col buffer lp2 at (step*512)/(step*512+128)
// Fused nonlinear epilogue writes 8 coefficient planes:
//   plane0 = kappa^2, plane1 = m1, plane2 = m2, plane3 = Hxx,
//   plane4 = Hxy, plane5 = Hyx, plane6 = Hyy, plane7 = D = 1/tau^2  (dt=1)
// Block: 256 threads = 8 wave32. Each block: one (b, h, 32-wide segment).
// Wave w: oc-tile = w&3 (16 output channels), pixel-group = w>>2 (16 pixels).
// ---------------------------------------------------------------------------
__global__ __launch_bounds__(256) void conv_theta_kernel(
    const float* __restrict__ x, const float* __restrict__ cw,
    float* __restrict__ planes)
{
  __shared__ float lw[64 * KTOT];      // weights [oc][k] (18 KB)
  __shared__ float lp[8 * 3 * 34];     // halo patch [ic][dh][dw] (3.2 KB)
  __shared__ float lp2[KTOT * 32];     // im2col [k][po] (9.2 KB)

  const int tid  = threadIdx.x;
  const int blk  = blockIdx.x;
  const int wseg = blk % NSEG;
  const int h    = (blk / NSEG) % Hn;
  const int b    = blk / (NSEG * Hn);
  const int w0   = wseg * 32;

  // stage weights: conv_w flat is already [oc*72 + k]
  for (int i = tid; i < 64 * KTOT; i += 256) lw[i] = cw[i];
  // stage halo patch: rows h-1..h+1, cols w0-1..w0+32, zero-padded (Dirichlet)
  for (int i = tid; i < 8 * 3 * 34; i += 256) {
    int ic = i / 102; int r = i - ic * 102; int dh = r / 34; int dw = r - dh * 34;
    int row = h + dh - 1; int col = w0 + dw - 1;
    float v = 0.0f;
    if (row >= 0 && row < Hn && col >= 0 && col < Wn)
      v = x[(((size_t)b * Tn + ic) * Hn + row) * Wn + col];
    lp[i] = v;
  }
  __syncthreads();

  // expand halo -> im2col layout lp2[k*32 + po] (k -> ic,kh,kw math done ONCE)
  for (int i = tid; i < KTOT * 32; i += 256) {
    int k  = i >> 5;
    int po = i & 31;
    int ic = k / 9; int r = k - ic * 9; int kh = r / 3; int kw = r - kh * 3;
    lp2[i] = lp[(ic * 3 + kh) * 34 + po + kw];
  }
  __syncthreads();

  const int wave    = tid >> 5;
  const int lane    = tid & 31;
  const int half    = lane >> 4;     // K-pair selector in A/B striping
  const int nl      = lane & 15;     // M for A operand, N for B operand
  const int oc_tile = wave & 3;
  const int pg      = wave >> 2;
  const int po      = pg * 16 + nl;  // pixel offset within 32-wide segment
  const int mrow    = oc_tile * 16 + nl;

  // loop-invariant per-lane bases; per-step offsets are immediates
  const v2f*   awp = (const v2f*)&lw[mrow * KTOT + half * 2]; // step -> index 2*step (16B)
  const float* bwp = &lp2[half * 64 + po];                    // step -> +128 floats

  v8f acc = {0.f, 0.f, 0.f, 0.f, 0.f, 0.f, 0.f, 0.f};

  #pragma unroll
  for (int step = 0; step < 18; ++step) {           // K = 72 = 18 * 4
    v2f av = awp[step * 2];                         // A[M][k0], A[M][k0+1]
    v2f bv;
    bv.x = bwp[step * 128];                         // B[k0][po]
    bv.y = bwp[step * 128 + 32];                    // B[k0+1][po]
    acc = __builtin_amdgcn_wmma_f32_16x16x4_f32(
        /*neg_a=*/false, av, /*neg_b=*/false, bv,
        /*c_mod=*/(short)0, acc, /*reuse_a=*/false, /*reuse_b=*/false);
  }

  // Epilogue: C/D layout -> VGPR v holds M=v (lanes 0-15) / M=v+8 (lanes 16-31)
  #pragma unroll
  for (int v = 0; v < 8; ++v) {
    const int M   = v + half * 8;
    const int oc  = oc_tile * 16 + M;
    const int grp = oc >> 3;
    const int t   = oc & 7;
    float val = acc[v];
    float outv;
    switch (grp) {
      case 0: { float s = csig(val, 0.01f, 1.0f); outv = s * s; } break;          // kappa^2
      case 1: case 2: outv = val; break;                                          // m1, m2
      case 3: case 6: outv = csig(val, 0.01f, 1.0f); break;                       // Hxx, Hyy
      case 4: case 5: outv = 0.1f * tanhf(val); break;                            // Hxy, Hyx
      default: { float tau = csig(val, 0.1f, 10.0f); outv = 1.0f / (tau * tau); } // D
        break;
    }
    planes[(size_t)grp * NP + ((((size_t)b * Tn + t) * Hn + h) * Wn + (w0 + po))] = outv;
  }
}

// ---------------------------------------------------------------------------
// 9-point operator A(u) with Dirichlet-zero boundaries.
// ---------------------------------------------------------------------------
__device__ __forceinline__ float applyA_at(
    const float* __restrict__ u, size_t base, int h, int w,
    float k2, float m1, float m2, float hxx, float hxy2, float hyy)
{
  size_t idx = base + (size_t)h * Wn + w;
  float uc = u[idx];
  float uE = (w < Wn - 1) ? u[idx + 1] : 0.f;
  float uW = (w > 0)      ? u[idx - 1] : 0.f;
  float uN = (h < Hn - 1) ? u[idx + Wn] : 0.f;
  float uS = (h > 0)      ? u[idx - Wn] : 0.f;
  float uNE = (h < Hn - 1 && w < Wn - 1) ? u[idx + Wn + 1] : 0.f;
  float uNW = (h < Hn - 1 && w > 0)      ? u[idx + Wn - 1] : 0.f;
  float uSE = (h > 0 && w < Wn - 1)      ? u[idx - Wn + 1] : 0.f;
  float uSW = (h > 0 && w > 0)           ? u[idx - Wn - 1] : 0.f;
  float dx  = 0.5f * (uE - uW);
  float dy  = 0.5f * (uN - uS);
  float dxx = uE + uW - 2.f * uc;
  float dyy = uN + uS - 2.f * uc;
  float dxy = 0.25f * (uNE - uNW - uSE + uSW);
  return k2 * uc + m1 * dx + m2 * dy - (hxx * dxx + hyy * dyy + hxy2 * dxy);
}

// Kernel 2: w1 = A(x)
__global__ __launch_bounds__(256) void stage1_kernel(
    const float* __restrict__ x, const float* __restrict__ planes,
    float* __restrict__ w1)
{
  int idx = blockIdx.x * 256 + threadIdx.x;
  int w = idx & (Wn - 1);
  int h = (idx >> 8) & (Hn - 1);
  int bt = idx >> 16;
  size_t base = (size_t)bt * Hn * Wn;
  float k2  = planes[(size_t)0 * NP + idx];
  float m1  = planes[(size_t)1 * NP + idx];
  float m2  = planes[(size_t)2 * NP + idx];
  float hxx = planes[(size_t)3 * NP + idx];
  float hxy = planes[(size_t)4 * NP + idx];
  float hyx = planes[(size_t)5 * NP + idx];
  float hyy = planes[(size_t)6 * NP + idx];
  w1[idx] = applyA_at(x, base, h, w, k2, m1, m2, hxx, hxy + hyx, hyy);
}

// Kernel 3: a2 = A(w1); Mx = x + a2; r = Mx - x[t-1];
// per-block deterministic tree reduction of Sum(D*r^2) and Sum(log D).
__global__ __launch_bounds__(256) void stage2_kernel(
    const float* __restrict__ x, const float* __restrict__ planes,
    const float* __restrict__ w1,
    float* __restrict__ part1, float* __restrict__ part2)
{
  __shared__ float s1[256];
  __shared__ float s2[256];
  int tid = threadIdx.x;
  int idx = blockIdx.x * 256 + tid;
  int w = idx & (Wn - 1);
  int h = (idx >> 8) & (Hn - 1);
  int bt = idx >> 16;
  size_t base = (size_t)bt * Hn * Wn;
  float k2  = planes[(size_t)0 * NP + idx];
  float m1  = planes[(size_t)1 * NP + idx];
  float m2  = planes[(size_t)2 * NP + idx];
  float hxx = planes[(size_t)3 * NP + idx];
  float hxy = planes[(size_t)4 * NP + idx];
  float hyx = planes[(size_t)5 * NP + idx];
  float hyy = planes[(size_t)6 * NP + idx];
  float Dv  = planes[(size_t)7 * NP + idx];

  float a2 = applyA_at(w1, base, h, w, k2, m1, m2, hxx, hxy + hyx, hyy);
  float mx = x[idx] + a2;                       // dt = 1
  int t = bt & 7;
  float xprev = (t > 0) ? x[idx - Hn * Wn] : 0.f;
  float r = mx - xprev;

  s1[tid] = Dv * r * r;
  s2[tid] = logf(Dv);
  __syncthreads();
  for (int s = 128; s > 0; s >>= 1) {
    if (tid < s) { s1[tid] += s1[tid + s]; s2[tid] += s2[tid + s]; }
    __syncthreads();
  }
  if (tid == 0) { part1[blockIdx.x] = s1[0]; part2[blockIdx.x] = s2[0]; }
}

// Kernel 4: fixed-order final reduction, out = 0.5*(xQx - logdetD)
__global__ __launch_bounds__(256) void finalize_kernel(
    const float* __restrict__ p1, const float* __restrict__ p2,
    float* __restrict__ out)
{
  __shared__ float s1[256];
  __shared__ float s2[256];
  int tid = threadIdx.x;
  float a = 0.f, b = 0.f;
  for (int i = tid; i < NBLK; i += 256) { a += p1[i]; b += p2[i]; }
  s1[tid] = a; s2[tid] = b;
  __syncthreads();
  for (int s = 128; s > 0; s >>= 1) {
    if (tid < s) { s1[tid] += s1[tid + s]; s2[tid] += s2[tid + s]; }
    __syncthreads();
  }
  if (tid == 0) out[0] = 0.5f * (s1[0] - s2[0]);
}

// ---------------------------------------------------------------------------
extern "C" void kernel_launch(void* const* d_in, const int* in_sizes, int n_in,
                              void* d_out, int out_size, void* d_ws, size_t ws_size,
                              hipStream_t stream)
{
  (void)in_sizes; (void)n_in; (void)out_size; (void)ws_size;
  const float* x  = (const float*)d_in[0];
  const float* cw = (const float*)d_in[1];
  float* ws     = (float*)d_ws;
  float* planes = ws;                         // 8 * NP floats (coefficient fields)
  float* w1     = ws + (size_t)8 * NP;        // NP floats (A(x))
  float* part1  = ws + (size_t)9 * NP;        // NBLK floats
  float* part2  = part1 + NBLK;               // NBLK floats

  conv_theta_kernel<<<NBLK, 256, 0, stream>>>(x, cw, planes);
  stage1_kernel  <<<NP / 256, 256, 0, stream>>>(x, planes, w1);
  stage2_kernel  <<<NP / 256, 256, 0, stream>>>(x, planes, w1, part1, part2);
  finalize_kernel<<<1, 256, 0, stream>>>(part1, part2, (float*)d_out);
}